// CausalGATLayer_81939386073360
// MI455X (gfx1250) — compile-verified
//
#include <hip/hip_runtime.h>
#include <math.h>

typedef __attribute__((ext_vector_type(2))) float v2f;
typedef __attribute__((ext_vector_type(8))) float v8f;

#define NN   512
#define IND  128
#define HIDD 256
#define OUTD 256
#define HD   64

__device__ __forceinline__ float elu_f(float x)  { return x > 0.f ? x : (expf(x) - 1.f); }
__device__ __forceinline__ float relu_f(float x) { return x > 0.f ? x : 0.f; }

// ---------------------------------------------------------------------------
// WMMA fp32 GEMM:  C(M,Nc) = act( X(M,K) * op(W) + bias ), optional accumulate.
// wtrans=1: op(W)=W^T, W is (Nc,K) row-major with leading dim ldw.
// wtrans=0: op(W)=W,  W is (K,Nc) row-major with leading dim ldw.
// One wave computes a 16x64 C tile: 4 accumulators share one A fragment per
// k-step -> 4 consecutive V_WMMA_F32_16X16X4_F32 per A load.
// A frag: lanes 0-15 hold rows m0..m0+15 at (k0,k0+1); lanes 16-31 at (k0+2,k0+3).
// B frag mirrored with columns in lanes. C/D: VGPR r -> row r (lanes 0-15) / r+8.
// Requires: M % 16 == 0, Nc % 64 == 0, K % 4 == 0 (true for this network).
// ---------------------------------------------------------------------------
__global__ void wmma_gemm_f32(const float* __restrict__ X, int ldx,
                              const float* __restrict__ W, int ldw,
                              const float* __restrict__ bias,
                              float* __restrict__ C, int ldc,
                              int M, int Ncols, int K,
                              int act, int accumulate, int wtrans)
{
    const int wave = threadIdx.x >> 5;
    const int lane = threadIdx.x & 31;
    const int lm   = lane & 15;
    const int hi   = lane >> 4;

    const int tN    = Ncols >> 6;               // 64-wide column tiles
    const int tiles = (M >> 4) * tN;
    const int tile  = blockIdx.x * 4 + wave;
    if (tile >= tiles) return;                  // whole-wave exit: EXEC all-1s for WMMA
    const int tm = tile / tN;
    const int tn = tile - tm * tN;
    const int row0 = tm << 4;
    const int col0 = tn << 6;

    union { v8f v; float f[8]; } acc[4];
    if (accumulate) {
        #pragma unroll
        for (int c = 0; c < 4; ++c)
            #pragma unroll
            for (int r = 0; r < 8; ++r)
                acc[c].f[r] = C[(size_t)(row0 + r + 8 * hi) * ldc + col0 + 16 * c + lm];
    } else {
        #pragma unroll
        for (int c = 0; c < 4; ++c)
            #pragma unroll
            for (int r = 0; r < 8; ++r) acc[c].f[r] = 0.f;
    }

    const float* xrow = X + (size_t)(row0 + lm) * ldx + 2 * hi;
    if (wtrans) {
        const float* wrow0 = W + (size_t)(col0 +  0 + lm) * ldw + 2 * hi;
        const float* wrow1 = W + (size_t)(col0 + 16 + lm) * ldw + 2 * hi;
        const float* wrow2 = W + (size_t)(col0 + 32 + lm) * ldw + 2 * hi;
        const float* wrow3 = W + (size_t)(col0 + 48 + lm) * ldw + 2 * hi;
        for (int k0 = 0; k0 < K; k0 += 4) {
            v2f a  = *(const v2f*)(xrow  + k0);
            v2f b0 = *(const v2f*)(wrow0 + k0);
            v2f b1 = *(const v2f*)(wrow1 + k0);
            v2f b2 = *(const v2f*)(wrow2 + k0);
            v2f b3 = *(const v2f*)(wrow3 + k0);
            acc[0].v = __builtin_amdgcn_wmma_f32_16x16x4_f32(false, a, false, b0, (short)0, acc[0].v, false, false);
            acc[1].v = __builtin_amdgcn_wmma_f32_16x16x4_f32(false, a, false, b1, (short)0, acc[1].v, false, false);
            acc[2].v = __builtin_amdgcn_wmma_f32_16x16x4_f32(false, a, false, b2, (short)0, acc[2].v, false, false);
            acc[3].v = __builtin_amdgcn_wmma_f32_16x16x4_f32(false, a, false, b3, (short)0, acc[3].v, false, false);
        }
    } else {
        for (int k0 = 0; k0 < K; k0 += 4) {
            v2f a = *(const v2f*)(xrow + k0);
            const float* wr0 = W + (size_t)(k0 + 2 * hi)     * ldw + col0 + lm;
            const float* wr1 = W + (size_t)(k0 + 2 * hi + 1) * ldw + col0 + lm;
            #pragma unroll
            for (int c = 0; c < 4; ++c) {
                v2f b;
                b.x = wr0[16 * c];
                b.y = wr1[16 * c];
                acc[c].v = __builtin_amdgcn_wmma_f32_16x16x4_f32(false, a, false, b, (short)0, acc[c].v, false, false);
            }
        }
    }

    #pragma unroll
    for (int c = 0; c < 4; ++c) {
        const int col = col0 + 16 * c + lm;
        const float bv = bias ? bias[col] : 0.f;
        #pragma unroll
        for (int r = 0; r < 8; ++r) {
            const int row = row0 + r + 8 * hi;
            float v = acc[c].f[r] + bv;
            if (act == 1) v = relu_f(v);
            else if (act == 2) v = elu_f(v);
            C[(size_t)row * ldc + col] = v;
        }
    }
}

// ---------------------------------------------------------------------------
// Head 1: per-node src/dst attention logits
// ---------------------------------------------------------------------------
__global__ void head1_srcdst(const float* __restrict__ Wh1,
                             const float* __restrict__ a_src,
                             const float* __restrict__ a_dst,
                             float* __restrict__ src1, float* __restrict__ dst1)
{
    int i = blockIdx.x * blockDim.x + threadIdx.x;
    if (i >= NN) return;
    const float* row = Wh1 + (size_t)i * HIDD;
    float s = 0.f, d = 0.f;
    for (int h = 0; h < HIDD; ++h) { float w = row[h]; s += w * a_src[h]; d += w * a_dst[h]; }
    src1[i] = s; dst1[i] = d;
}

// adj -> float matrix + row degrees (for uniform-softmax heads 3/4)
__global__ void adj_to_float_deg(const int* __restrict__ adj,
                                 float* __restrict__ adjf, float* __restrict__ deg)
{
    __shared__ int red[256];
    const int row = blockIdx.x, tid = threadIdx.x;
    int cnt = 0;
    for (int j = tid; j < NN; j += 256) {
        int a = adj[(size_t)row * NN + j];
        adjf[(size_t)row * NN + j] = (float)a;
        cnt += a;
    }
    red[tid] = cnt; __syncthreads();
    for (int s = 128; s > 0; s >>= 1) { if (tid < s) red[tid] += red[tid + s]; __syncthreads(); }
    if (tid == 0) deg[row] = (float)red[0];
}

// Head 1 masked softmax row: alpha[i,:] = softmax_j( leaky(src_i+dst_j+0.1(d_i+d_j)) | adj )
__global__ void head1_alpha(const int* __restrict__ adj,
                            const float* __restrict__ src1, const float* __restrict__ dst1,
                            const float* __restrict__ delta, float* __restrict__ alpha)
{
    __shared__ float red[256];
    const int i = blockIdx.x, tid = threadIdx.x;
    const float si = src1[i], di = delta[i];
    const int j0 = tid, j1 = tid + 256;
    const int a0 = adj[(size_t)i * NN + j0];
    const int a1 = adj[(size_t)i * NN + j1];
    float s0 = si + dst1[j0] + 0.1f * (di + delta[j0]);
    float s1 = si + dst1[j1] + 0.1f * (di + delta[j1]);
    s0 = s0 > 0.f ? s0 : 0.2f * s0;
    s1 = s1 > 0.f ? s1 : 0.2f * s1;
    const float e0 = a0 ? s0 : -INFINITY;
    const float e1 = a1 ? s1 : -INFINITY;
    red[tid] = fmaxf(e0, e1); __syncthreads();
    for (int s = 128; s > 0; s >>= 1) { if (tid < s) red[tid] = fmaxf(red[tid], red[tid + s]); __syncthreads(); }
    const float m = red[0]; __syncthreads();
    const float p0 = a0 ? expf(e0 - m) : 0.f;
    const float p1 = a1 ? expf(e1 - m) : 0.f;
    red[tid] = p0 + p1; __syncthreads();
    for (int s = 128; s > 0; s >>= 1) { if (tid < s) red[tid] += red[tid + s]; __syncthreads(); }
    const float denom = red[0];
    alpha[(size_t)i * NN + j0] = p0 / denom;
    alpha[(size_t)i * NN + j1] = p1 / denom;
}

// ---------------------------------------------------------------------------
// Head 2 pair sum: part[b,h] = sum over i in block-chunk, j!=i of relu(A[i,h]+B[j,h]+b1[h])
// 64 rows of A staged in 64KB LDS; thread = hidden channel h.
// ---------------------------------------------------------------------------
__global__ void pair_sum(const float* __restrict__ A2, const float* __restrict__ B2,
                         const float* __restrict__ b1, float* __restrict__ part)
{
    __shared__ float As[64 * HIDD];               // 64 KB
    const int i0 = blockIdx.x * 64;
    const int tid = threadIdx.x;                  // 256 threads == HID channels
    for (int idx = tid; idx < 64 * HIDD; idx += 256)
        As[idx] = A2[(size_t)i0 * HIDD + idx];
    __syncthreads();
    const int h = tid;
    const float bh = b1[h];
    float sum = 0.f;
    for (int j = 0; j < NN; ++j) {
        const float t = B2[(size_t)j * HIDD + h] + bh;
        #pragma unroll 8
        for (int ii = 0; ii < 64; ++ii)
            sum += relu_f(As[ii * HIDD + h] + t);
    }
    for (int ii = 0; ii < 64; ++ii)               // remove diagonal i==j terms
        sum -= relu_f(As[ii * HIDD + h] + B2[(size_t)(i0 + ii) * HIDD + h] + bh);
    part[blockIdx.x * HIDD + h] = sum;
}

// Reduce partials -> S, then causal_mean = ce_w2 @ (S/(N^2-N)) + b2, h2row = hp_w1 @ cm + hp_b1
__global__ void head2_finish(const float* __restrict__ part,
                             const float* __restrict__ ce_w2, const float* __restrict__ ce_b2,
                             const float* __restrict__ hp_w,  const float* __restrict__ hp_b,
                             float* __restrict__ h2row)
{
    __shared__ float S[HIDD];
    __shared__ float cm[HD];
    const int tid = threadIdx.x;                  // 256 threads
    float s = 0.f;
    for (int b = 0; b < 8; ++b) s += part[b * HIDD + tid];
    S[tid] = s;
    __syncthreads();
    if (tid < HD) {
        float acc = 0.f;
        for (int hh = 0; hh < HIDD; ++hh) acc += ce_w2[tid * HIDD + hh] * S[hh];
        cm[tid] = acc * (1.f / (float)(NN * NN - NN)) + ce_b2[tid];
    }
    __syncthreads();
    if (tid < HD) {
        float acc = 0.f;
        for (int q = 0; q < HD; ++q) acc += hp_w[HD * HD + tid * HD + q] * cm[q];
        h2row[tid] = acc + hp_b[HD + tid];
    }
}

__global__ void fill_h2(const float* __restrict__ h2row, float* __restrict__ Hc)
{
    int idx = blockIdx.x * 256 + threadIdx.x;
    if (idx >= NN * HD) return;
    int i = idx >> 6, k = idx & 63;
    Hc[(size_t)i * OUTD + HD + k] = h2row[k];
}

__global__ void scale_rows(float* __restrict__ x, const float* __restrict__ deg)
{
    int idx = blockIdx.x * 256 + threadIdx.x;
    if (idx >= NN * HD) return;
    x[idx] /= deg[idx >> 6];
}

// eLU + LayerNorm epilogue, one row per block
__global__ void elu_layernorm(const float* __restrict__ fused,
                              const float* __restrict__ g, const float* __restrict__ b,
                              float* __restrict__ out)
{
    __shared__ float red[OUTD];
    const int i = blockIdx.x, c = threadIdx.x;
    const float e = elu_f(fused[(size_t)i * OUTD + c]);
    red[c] = e; __syncthreads();
    for (int s = 128; s > 0; s >>= 1) { if (c < s) red[c] += red[c + s]; __syncthreads(); }
    const float mu = red[0] * (1.f / OUTD); __syncthreads();
    const float d = e - mu;
    red[c] = d * d; __syncthreads();
    for (int s = 128; s > 0; s >>= 1) { if (c < s) red[c] += red[c + s]; __syncthreads(); }
    const float var = red[0] * (1.f / OUTD);
    out[(size_t)i * OUTD + c] = d / sqrtf(var + 1e-5f) * g[c] + b[c];
}

// ---------------------------------------------------------------------------
extern "C" void kernel_launch(void* const* d_in, const int* in_sizes, int n_in,
                              void* d_out, int out_size, void* d_ws, size_t ws_size,
                              hipStream_t stream)
{
    (void)in_sizes; (void)n_in; (void)out_size; (void)ws_size;
    const float* V     = (const float*)d_in[0];
    const int*   adj   = (const int*)  d_in[1];
    const float* delta = (const float*)d_in[2];
    const float* prevh = (const float*)d_in[3];
    const float* W1    = (const float*)d_in[4];
    const float* a_src = (const float*)d_in[5];
    const float* a_dst = (const float*)d_in[6];
    const float* ce_w1 = (const float*)d_in[7];   // (256, 256) = [W_L | W_R]
    const float* ce_b1 = (const float*)d_in[8];
    const float* ce_w2 = (const float*)d_in[9];
    const float* ce_b2 = (const float*)d_in[10];
    const float* te_w1 = (const float*)d_in[11];  // (256, 192)
    const float* te_b1 = (const float*)d_in[12];
    const float* te_w2 = (const float*)d_in[13];
    const float* te_b2 = (const float*)d_in[14];
    // d_in[15] = ta_w: row-constant scores -> uniform softmax, value unused
    const float* se_w1 = (const float*)d_in[16];
    const float* se_b1 = (const float*)d_in[17];
    const float* se_w2 = (const float*)d_in[18];
    const float* se_b2 = (const float*)d_in[19];
    // d_in[20] = sa_w: unused for same reason
    const float* hp_w  = (const float*)d_in[21];  // (4,64,64)
    const float* hp_b  = (const float*)d_in[22];  // (4,64)
    const float* hf_w1 = (const float*)d_in[23];
    const float* hf_b1 = (const float*)d_in[24];
    const float* hf_w2 = (const float*)d_in[25];
    const float* hf_b2 = (const float*)d_in[26];
    const float* ln_g  = (const float*)d_in[27];
    const float* ln_b  = (const float*)d_in[28];
    float* out = (float*)d_out;

    float* ws = (float*)d_ws;
    float* Wh1   = ws; ws += NN * HIDD;
    float* adjf  = ws; ws += NN * NN;
    float* alpha = ws; ws += NN * NN;
    float* H1    = ws; ws += NN * HIDD;
    float* A2    = ws; ws += NN * HIDD;
    float* B2    = ws; ws += NN * HIDD;
    float* th    = ws; ws += NN * HIDD;   // reused: temporal hidden, then situation hidden
    float* tfeat = ws; ws += NN * HD;
    float* sfeat = ws; ws += NN * HD;
    float* nsum3 = ws; ws += NN * HD;
    float* nsum4 = ws; ws += NN * HD;
    float* Hc    = ws; ws += NN * OUTD;
    float* F1    = ws; ws += NN * OUTD;
    float* fused = ws; ws += NN * OUTD;
    float* src1  = ws; ws += NN;
    float* dst1  = ws; ws += NN;
    float* deg   = ws; ws += NN;
    float* part  = ws; ws += 8 * HIDD;
    float* h2row = ws; ws += HD;

    auto gemm = [&](const float* X, int ldx, const float* W, int ldw, const float* bias,
                    float* C, int ldc, int M, int Nc, int K, int act, int accum, int wtrans) {
        int tiles = (M / 16) * (Nc / 64);
        int blocks = (tiles + 3) / 4;
        wmma_gemm_f32<<<blocks, 128, 0, stream>>>(X, ldx, W, ldw, bias, C, ldc,
                                                  M, Nc, K, act, accum, wtrans);
    };

    // shared precursors
    adj_to_float_deg<<<NN, 256, 0, stream>>>(adj, adjf, deg);

    // ---- Head 1 ----
    gemm(V, IND, W1, IND, nullptr, Wh1, HIDD, NN, HIDD, IND, 0, 0, 1);        // Wh1 = V@W1^T
    head1_srcdst<<<(NN + 127) / 128, 128, 0, stream>>>(Wh1, a_src, a_dst, src1, dst1);
    head1_alpha<<<NN, 256, 0, stream>>>(adj, src1, dst1, delta, alpha);
    gemm(alpha, NN, Wh1, HIDD, nullptr, H1, HIDD, NN, HIDD, NN, 2, 0, 0);     // H1 = elu(alpha@Wh1)
    gemm(H1, HIDD, hp_w + 0 * HD * HD, HD, hp_b + 0 * HD,
         Hc + 0, OUTD, NN, HD, HD, 0, 0, 1);                                  // h1 -> Hc[:,0:64]

    // ---- Head 2 (rank-structured pair MLP) ----
    gemm(V, IND, ce_w1, 2 * IND, nullptr, A2, HIDD, NN, HIDD, IND, 0, 0, 1);        // A = V@W_L^T
    gemm(V, IND, ce_w1 + IND, 2 * IND, nullptr, B2, HIDD, NN, HIDD, IND, 0, 0, 1);  // B = V@W_R^T
    pair_sum<<<8, 256, 0, stream>>>(A2, B2, ce_b1, part);
    head2_finish<<<1, 256, 0, stream>>>(part, ce_w2, ce_b2, hp_w, hp_b, h2row);
    fill_h2<<<(NN * HD + 255) / 256, 256, 0, stream>>>(h2row, Hc);            // h2 -> Hc[:,64:128]

    // ---- Head 3 (temporal; uniform softmax == adj row-mean) ----
    gemm(V, IND, te_w1, IND + HD, nullptr, th, HIDD, NN, HIDD, IND, 0, 0, 1);
    gemm(prevh, HD, te_w1 + IND, IND + HD, te_b1, th, HIDD, NN, HIDD, HD, 1, 1, 1); // relu(tin@te_w1^T+b)
    gemm(th, HIDD, te_w2, HIDD, te_b2, tfeat, HD, NN, HD, HIDD, 0, 0, 1);     // tfeat
    gemm(adjf, NN, tfeat, HD, nullptr, nsum3, HD, NN, HD, NN, 0, 0, 0);       // adj @ tfeat
    scale_rows<<<(NN * HD + 255) / 256, 256, 0, stream>>>(nsum3, deg);
    gemm(nsum3, HD, hp_w + 2 * HD * HD, HD, hp_b + 2 * HD,
         Hc + 2 * HD, OUTD, NN, HD, HD, 0, 0, 1);                             // h3 -> Hc[:,128:192]

    // ---- Head 4 (situation; same uniform-softmax structure) ----
    gemm(V, IND, se_w1, IND, se_b1, th, HIDD, NN, HIDD, IND, 1, 0, 1);        // relu(V@se_w1^T+b)
    gemm(th, HIDD, se_w2, HIDD, se_b2, sfeat, HD, NN, HD, HIDD, 0, 0, 1);     // sfeat
    gemm(adjf, NN, sfeat, HD, nullptr, nsum4, HD, NN, HD, NN, 0, 0, 0);       // adj @ sfeat
    scale_rows<<<(NN * HD + 255) / 256, 256, 0, stream>>>(nsum4, deg);
    gemm(nsum4, HD, hp_w + 3 * HD * HD, HD, hp_b + 3 * HD,
         Hc + 3 * HD, OUTD, NN, HD, HD, 0, 0, 1);                             // h4 -> Hc[:,192:256]

    // ---- Fuse + eLU + LayerNorm ----
    gemm(Hc, OUTD, hf_w1, OUTD, hf_b1, F1, OUTD, NN, OUTD, OUTD, 1, 0, 1);    // relu(Hc@hf_w1^T+b1)
    gemm(F1, OUTD, hf_w2, OUTD, hf_b2, fused, OUTD, NN, OUTD, OUTD, 0, 0, 1); // fused
    elu_layernorm<<<NN, 256, 0, stream>>>(fused, ln_g, ln_b, out);
}